// RPN_16329465660238
// MI455X (gfx1250) — compile-verified
//
#include <hip/hip_runtime.h>
#include <hip/hip_bf16.h>

#define AA    3
#define HH    200
#define WWID  336
#define HWW   (HH * WWID)      // 67200
#define AHW   (AA * HH * WWID) // 201600
#define PRE_N 2000
#define POST_N 1000
#define SORTN 2048
#define NMS_THR_C 0.7f
#define MAX_OFF 4.135166556742356f   // log(1000/16)
#define NMS_DEPTH 4                  // async pipeline depth (power of 2)

__device__ __forceinline__ unsigned fkey(float f) {
    unsigned u = __float_as_uint(f);
    return (u & 0x80000000u) ? ~u : (u | 0x80000000u);
}
__device__ __forceinline__ float key_to_val(unsigned k) {
    unsigned u = (k & 0x80000000u) ? (k & 0x7FFFFFFFu) : ~k;
    return __uint_as_float(u);
}

// CDNA5 async global->LDS copy (ASYNCcnt-tracked). One b64 per lane.
// 'dep' is a dummy data operand that forces row consumption to be ordered
// before the LDS slot is reused by the DMA.
__device__ __forceinline__ void async_row_b64(unsigned lds_off,
                                              unsigned long long gaddr,
                                              unsigned dep) {
    asm volatile("global_load_async_to_lds_b64 %0, %1, off"
                 :
                 : "v"(lds_off), "v"(gaddr), "v"(dep)
                 : "memory");
}
template <int Nc>
__device__ __forceinline__ void wait_asynccnt() {
    asm volatile("s_wait_asynccnt %0" : : "i"(Nc) : "memory");
}

// ---------------------------------------------------------------------------
// Kernel 1: per-batch exact top-2000 (3-level radix select + bitonic sort on
// composite (key, ~index) so lax.top_k tie-breaking is reproduced exactly),
// then gather + decode + clip, writing boxes/scores to workspace.
// ---------------------------------------------------------------------------
__global__ __launch_bounds__(1024)
void rpn_topk_decode(const float* __restrict__ logits,
                     const float* __restrict__ regs,
                     const float* __restrict__ anchors,
                     const int* __restrict__ sizes,
                     float* __restrict__ wsBoxes,
                     float* __restrict__ wsScores) {
    __shared__ unsigned long long sel[SORTN];   // 16 KB
    __shared__ unsigned hist[2048];             // 8 KB (reused across levels)
    __shared__ unsigned sAcc;
    __shared__ int sB1, sB2, sSelCount;
    __shared__ unsigned sThresh;

    const int n   = blockIdx.x;
    const int tid = threadIdx.x;
    const int nt  = blockDim.x;
    const float* lg = logits + (size_t)n * AHW;

    // ---- level 1: top 11 bits ----
    for (int b = tid; b < 2048; b += nt) hist[b] = 0u;
    __syncthreads();
    for (int e = tid; e < AHW; e += nt) {
        __builtin_prefetch(lg + e + nt, 0, 1);   // global_prefetch_b8
        unsigned k = fkey(lg[e]);
        atomicAdd(&hist[k >> 21], 1u);
    }
    __syncthreads();
    if (tid == 0) {
        unsigned acc = 0; int b;
        for (b = 2047; b > 0; --b) {
            unsigned h = hist[b];
            if (acc + h >= PRE_N) break;
            acc += h;
        }
        sB1 = b; sAcc = acc;
    }
    __syncthreads();
    const unsigned B1 = (unsigned)sB1;

    // ---- level 2: next 11 bits, restricted to bucket B1 ----
    __syncthreads();
    for (int b = tid; b < 2048; b += nt) hist[b] = 0u;
    __syncthreads();
    for (int e = tid; e < AHW; e += nt) {
        __builtin_prefetch(lg + e + nt, 0, 1);
        unsigned k = fkey(lg[e]);
        if ((k >> 21) == B1) atomicAdd(&hist[(k >> 10) & 2047u], 1u);
    }
    __syncthreads();
    if (tid == 0) {
        unsigned acc = sAcc; int b;
        for (b = 2047; b > 0; --b) {
            unsigned h = hist[b];
            if (acc + h >= PRE_N) break;
            acc += h;
        }
        sB2 = b; sAcc = acc;
    }
    __syncthreads();
    const unsigned B2 = (unsigned)sB2;
    const unsigned pref22 = (B1 << 11) | B2;

    // ---- level 3: last 10 bits, restricted to 22-bit prefix ----
    __syncthreads();
    for (int b = tid; b < 2048; b += nt) hist[b] = 0u;
    __syncthreads();
    for (int e = tid; e < AHW; e += nt) {
        __builtin_prefetch(lg + e + nt, 0, 1);
        unsigned k = fkey(lg[e]);
        if ((k >> 10) == pref22) atomicAdd(&hist[k & 1023u], 1u);
    }
    __syncthreads();
    if (tid == 0) {
        unsigned acc = sAcc; int b;
        for (b = 1023; b > 0; --b) {
            unsigned h = hist[b];
            if (acc + h >= PRE_N) break;
            acc += h;
        }
        sThresh = (B1 << 21) | (B2 << 10) | (unsigned)b;
        sSelCount = 0;
    }
    __syncthreads();
    const unsigned Kstar = sThresh;

    // ---- compact key >= Kstar (count in [2000, ~2048)) ----
    for (int e = tid; e < AHW; e += nt) {
        __builtin_prefetch(lg + e + nt, 0, 1);
        unsigned k = fkey(lg[e]);
        if (k >= Kstar) {
            int a = e / HWW;
            int r = e - a * HWW;
            unsigned i = (unsigned)(r * AA + a);   // score-space index
            int p = atomicAdd(&sSelCount, 1);
            if (p < SORTN)
                sel[p] = ((unsigned long long)k << 32) | (unsigned)(~i);
        }
    }
    __syncthreads();
    int cnt = sSelCount; if (cnt > SORTN) cnt = SORTN;
    for (int p = cnt + tid; p < SORTN; p += nt) sel[p] = 0ull;
    __syncthreads();

    // ---- bitonic sort, descending; equal keys order by ascending index ----
    for (unsigned kk = 2; kk <= SORTN; kk <<= 1) {
        for (unsigned j = kk >> 1; j > 0; j >>= 1) {
            for (unsigned t = tid; t < SORTN; t += nt) {
                unsigned p = t ^ j;
                if (p > t) {
                    unsigned long long a = sel[t], b = sel[p];
                    bool up = (t & kk) == 0;
                    if (up ? (a < b) : (a > b)) { sel[t] = b; sel[p] = a; }
                }
            }
            __syncthreads();
        }
    }

    // ---- gather / decode / clip top PRE_N ----
    const float fh = (float)sizes[2 * n + 0];
    const float fw = (float)sizes[2 * n + 1];
    for (int t = tid; t < PRE_N; t += nt) {
        unsigned long long s = sel[t];
        unsigned k = (unsigned)(s >> 32);
        unsigned i = ~(unsigned)(s & 0xFFFFFFFFull);
        float logit = key_to_val(k);
        float score = 1.f / (1.f + expf(-logit));

        int a  = (int)(i % AA);
        int hw = (int)(i / AA);
        int w  = hw % WWID;
        int h  = hw / WWID;

        const float* anc = anchors + (((size_t)n * AHW + i) << 2);
        float ax1 = anc[0], ay1 = anc[1], ax2 = anc[2], ay2 = anc[3];

        size_t rbase = (((size_t)n * (AA * 4) + (size_t)(a * 4)) * HH + h) * WWID + w;
        float dx = regs[rbase];
        float dy = regs[rbase + (size_t)HWW];
        float dw = fminf(regs[rbase + 2 * (size_t)HWW], MAX_OFF);
        float dh = fminf(regs[rbase + 3 * (size_t)HWW], MAX_OFF);

        float ws_ = ax2 - ax1 + 1.f, hs_ = ay2 - ay1 + 1.f;
        float xc = ax1 + 0.5f * ws_ + dx * ws_;
        float yc = ay1 + 0.5f * hs_ + dy * hs_;
        float w2 = ws_ * expf(dw);
        float h2 = hs_ * expf(dh);
        float x1 = xc - 0.5f * w2;
        float y1 = yc - 0.5f * h2;
        float x2 = xc + 0.5f * w2 - 1.f;
        float y2 = yc + 0.5f * h2 - 1.f;
        x1 = fminf(fmaxf(x1, 0.f), fw - 1.f);
        y1 = fminf(fmaxf(y1, 0.f), fh - 1.f);
        x2 = fminf(fmaxf(x2, 0.f), fw - 1.f);
        y2 = fminf(fmaxf(y2, 0.f), fh - 1.f);

        float4* bo = (float4*)(wsBoxes + (((size_t)n * PRE_N + t) << 2));
        *bo = make_float4(x1, y1, x2, y2);
        wsScores[(size_t)n * PRE_N + t] = score;
    }
}

// ---------------------------------------------------------------------------
// Kernel 2: fully parallel suppression bitmask. mask[n][i][w] holds bits for
// j in [32w, 32w+32): 1 iff j > i and IoU(i,j) > 0.7. One thread per word.
// ---------------------------------------------------------------------------
__global__ __launch_bounds__(256)
void rpn_mask(const float* __restrict__ wsBoxes,
              unsigned* __restrict__ mask, int N) {
    int gid = blockIdx.x * blockDim.x + threadIdx.x;
    int total = N * PRE_N * 64;
    if (gid >= total) return;
    int n = gid / (PRE_N * 64);
    int r = gid - n * (PRE_N * 64);
    int i = r >> 6;
    int w = r & 63;
    int jbase = w * 32;
    unsigned m = 0u;
    if (jbase < PRE_N && jbase + 31 > i) {
        const float4* B = (const float4*)(wsBoxes + ((size_t)n * PRE_N << 2));
        float4 bi = B[i];
        float areai = (bi.z - bi.x + 1.f) * (bi.w - bi.y + 1.f);
        int j0 = (jbase > i + 1) ? jbase : (i + 1);
        int j1 = (jbase + 32 < PRE_N) ? (jbase + 32) : PRE_N;
        for (int j = j0; j < j1; ++j) {
            float4 bj = B[j];
            float xtl = fmaxf(bi.x, bj.x);
            float ytl = fmaxf(bi.y, bj.y);
            float xbr = fminf(bi.z, bj.z);
            float ybr = fminf(bi.w, bj.w);
            float iw = fmaxf(xbr - xtl + 1.f, 0.f);
            float ih = fmaxf(ybr - ytl + 1.f, 0.f);
            float inter = iw * ih;
            float areaj = (bj.z - bj.x + 1.f) * (bj.w - bj.y + 1.f);
            float iou = inter / (areai + areaj - inter);
            if (iou > NMS_THR_C) m |= (1u << (j - jbase));
        }
    }
    mask[(size_t)gid] = m;
}

// ---------------------------------------------------------------------------
// Kernel 3: serial greedy NMS per batch on one wave32. Alive bitset lives in
// registers (2 words/lane), current bit broadcast with __shfl. Mask rows are
// streamed 4 deep into an LDS ring with global_load_async_to_lds_b64
// (ASYNCcnt pipeline; async loads complete in order, so s_wait_asynccnt 3
// guarantees the oldest row is resident). Popcount prefix ranking then
// reproduces the post-NMS top-k ordering and writes (N,1000,5).
// ---------------------------------------------------------------------------
__global__ __launch_bounds__(1024)
void rpn_nms_out(const float* __restrict__ wsBoxes,
                 const float* __restrict__ wsScores,
                 const unsigned* __restrict__ mask,
                 float* __restrict__ out) {
    __shared__ __align__(16) unsigned ring[NMS_DEPTH * 64];  // 4 rows x 256 B
    __shared__ unsigned aliveS[64];
    __shared__ int wpref[65];
    const int n = blockIdx.x;
    const int tid = threadIdx.x;

    if (tid < 32) {
        const int l = tid;
        // lane l owns alive words 2l (bits 64l..64l+31) and 2l+1
        int base0 = l * 64, base1 = l * 64 + 32;
        unsigned ax = (base0 >= PRE_N) ? 0u
                     : ((PRE_N - base0 >= 32) ? 0xFFFFFFFFu : ((1u << (PRE_N - base0)) - 1u));
        unsigned ay = (base1 >= PRE_N) ? 0u
                     : ((PRE_N - base1 >= 32) ? 0xFFFFFFFFu : ((1u << (PRE_N - base1)) - 1u));

        const unsigned* mrow = mask + (size_t)n * PRE_N * 64;
        const unsigned ring_base = (unsigned)(uintptr_t)(&ring[0]);
        const unsigned my_lds = ring_base + 8u * (unsigned)l;

        // prime the pipeline: rows 0..NMS_DEPTH-1 in flight
        for (int d = 0; d < NMS_DEPTH; ++d)
            async_row_b64(my_lds + (unsigned)(d * 256),
                          (unsigned long long)(uintptr_t)(mrow + (size_t)d * 64 + 2 * l),
                          0u);

        for (int i = 0; i < PRE_N; ++i) {
            wait_asynccnt<NMS_DEPTH - 1>();          // row i resident in LDS
            const int slot = i & (NMS_DEPTH - 1);
            uint2 rr = *(const uint2*)(&ring[slot * 64 + 2 * l]);

            int src = i >> 6;
            unsigned myw = ((i >> 5) & 1) ? ay : ax;
            unsigned wv = __shfl(myw, src, 32);
            unsigned keepm = ((wv >> (i & 31)) & 1u) ? 0xFFFFFFFFu : 0u;  // branchless
            ax &= ~(rr.x & keepm);
            ay &= ~(rr.y & keepm);

            int nx = i + NMS_DEPTH;
            if (nx < PRE_N)
                async_row_b64(my_lds + (unsigned)(slot * 256),
                              (unsigned long long)(uintptr_t)(mrow + (size_t)nx * 64 + 2 * l),
                              ax /* dep: row i consumed before slot reuse */);
        }
        wait_asynccnt<0>();

        aliveS[2 * l]     = ax;
        aliveS[2 * l + 1] = ay;
    }
    __syncthreads();
    if (tid == 0) {
        int acc = 0;
        for (int w = 0; w < 64; ++w) { wpref[w] = acc; acc += __popc(aliveS[w]); }
        wpref[64] = acc;
    }
    __syncthreads();
    const int S = wpref[64];
    for (int i = tid; i < PRE_N; i += blockDim.x) {
        unsigned word = aliveS[i >> 5];
        bool keep = (word >> (i & 31)) & 1u;
        int pe = wpref[i >> 5] + __popc(word & ((1u << (i & 31)) - 1u));
        int rank = keep ? pe : (S + (i - pe));   // survivors first, then -1s by index
        if (rank < POST_N) {
            const float4 b = *(const float4*)(wsBoxes + (((size_t)n * PRE_N + i) << 2));
            float* o = out + ((size_t)n * POST_N + rank) * 5;
            o[0] = b.x; o[1] = b.y; o[2] = b.z; o[3] = b.w;
            o[4] = keep ? wsScores[(size_t)n * PRE_N + i] : -1.0f;
        }
    }
}

extern "C" void kernel_launch(void* const* d_in, const int* in_sizes, int n_in,
                              void* d_out, int out_size, void* d_ws, size_t ws_size,
                              hipStream_t stream) {
    const float* logits  = (const float*)d_in[0];
    const float* regs    = (const float*)d_in[1];
    const float* anchors = (const float*)d_in[2];
    const int*   sizes   = (const int*)d_in[3];
    float* out = (float*)d_out;

    const int N = in_sizes[0] / AHW;

    char* ws = (char*)d_ws;
    float*    wsBoxes  = (float*)ws;                                        // N*2000*4 f32
    float*    wsScores = (float*)(ws + (size_t)N * PRE_N * 4 * sizeof(float));
    unsigned* wsMask   = (unsigned*)(ws + (size_t)N * PRE_N * 5 * sizeof(float)); // N*2000*64 u32

    rpn_topk_decode<<<N, 1024, 0, stream>>>(logits, regs, anchors, sizes,
                                            wsBoxes, wsScores);

    int totalMask = N * PRE_N * 64;
    rpn_mask<<<(totalMask + 255) / 256, 256, 0, stream>>>(wsBoxes, wsMask, N);

    rpn_nms_out<<<N, 1024, 0, stream>>>(wsBoxes, wsScores, wsMask, out);
}